// ResidualGATBlock_32779190403595
// MI455X (gfx1250) — compile-verified
//
#include <hip/hip_runtime.h>
#include <hip/hip_bf16.h>

// ---------------- problem constants (from reference) ----------------
#define N_NODES 50000
#define IN_FEAT 128
#define HC      64          // HEADS*OUT_F = 4*16
#define HEADS   4
#define OUT_F   16
#define E_EDGES 800000
#define E_TOT   (E_EDGES + N_NODES)   // self-loops appended
#define BN_EPS  1e-5f
#define NEG_SLOPE 0.2f

#define USE_ASYNC_LDS 1     // gfx1250 GLOBAL_LOAD_ASYNC_TO_LDS_B128 staging of weights

typedef __attribute__((ext_vector_type(2))) float v2f;
typedef __attribute__((ext_vector_type(8))) float v8f;

// order-preserving float<->uint encoding for atomicMax on floats
__device__ __forceinline__ unsigned int fenc(float f) {
  unsigned int u = __float_as_uint(f);
  return (u & 0x80000000u) ? ~u : (u | 0x80000000u);
}
__device__ __forceinline__ float fdec(unsigned int u) {
  return __uint_as_float((u & 0x80000000u) ? (u & 0x7FFFFFFFu) : ~u);
}

// ---------------------------------------------------------------
// Kernel 0: init scratch (scatter accum, denom, encoded -inf max, BN sums)
// ---------------------------------------------------------------
__global__ void init_kernel(float4* __restrict__ sout4, float* __restrict__ denom,
                            unsigned int* __restrict__ emax, float* __restrict__ sums) {
  long long idx = (long long)blockIdx.x * blockDim.x + threadIdx.x;
  if (idx < (long long)N_NODES * (HC / 4))
    sout4[idx] = make_float4(0.f, 0.f, 0.f, 0.f);
  if (idx < (long long)N_NODES * HEADS) {
    denom[idx] = 0.0f;
    emax[idx]  = 0x007FFFFFu;   // fenc(-inf)
  }
  if (idx < 2 * HC) sums[idx] = 0.0f;  // sums[0:64], sumsq[64:128] contiguous
}

// ---------------------------------------------------------------
// Kernel 1: fused dual GEMM  h = x@W  and  res = x@W_res
// V_WMMA_F32_16X16X4_F32, weights staged in LDS (async direct-to-LDS copy),
// one 16-row block per wave, 4+4 output tiles.
// A 16x4 f32 layout: lanes 0-15 hold K=0,1 ; lanes 16-31 hold K=2,3 (per lane M = lane&15)
// B 4x16 f32 layout: vgpr0 = K(0|2), vgpr1 = K(1|3), N = lane&15
// C/D 16x16 f32: vgpr v -> row v + 8*(lane>=16), col = lane&15
// ---------------------------------------------------------------
__global__ void __launch_bounds__(256)
gemm_dual_kernel(const float* __restrict__ x, const float* __restrict__ W,
                 const float* __restrict__ Wres, float* __restrict__ h,
                 float* __restrict__ res) {
  __shared__ float sW [IN_FEAT * HC];
  __shared__ float sWr[IN_FEAT * HC];
  const int t = threadIdx.x;

#if USE_ASYNC_LDS
  {
    // 2048 float4 per matrix / 256 threads = 8 async b128 copies each
    const unsigned ldsW  = (unsigned)(size_t)(&sW[0]);
    const unsigned ldsWr = (unsigned)(size_t)(&sWr[0]);
    for (int i = t; i < (IN_FEAT * HC) / 4; i += 256) {
      unsigned long long gw  = (unsigned long long)(size_t)(W    + i * 4);
      unsigned long long gwr = (unsigned long long)(size_t)(Wres + i * 4);
      unsigned lw  = ldsW  + i * 16;
      unsigned lwr = ldsWr + i * 16;
      asm volatile("global_load_async_to_lds_b128 %0, %1, off"
                   :: "v"(lw), "v"(gw) : "memory");
      asm volatile("global_load_async_to_lds_b128 %0, %1, off"
                   :: "v"(lwr), "v"(gwr) : "memory");
    }
    asm volatile("s_wait_asynccnt 0x0" ::: "memory");
  }
#else
  for (int i = t; i < IN_FEAT * HC; i += 256) {
    sW[i]  = W[i];
    sWr[i] = Wres[i];
  }
#endif
  __syncthreads();

  const int wave     = t >> 5;
  const int lane     = t & 31;
  const int rowBlock = blockIdx.x * 8 + wave;      // one 16-row block per wave
  if (rowBlock * 16 >= N_NODES) return;            // wave-uniform exit, EXEC stays full
  const int row0  = rowBlock * 16;
  const int m     = lane & 15;
  const int khalf = (lane >> 4) << 1;              // 0 or 2

  v8f accW[4], accR[4];
#pragma unroll
  for (int i = 0; i < 4; ++i) { accW[i] = (v8f)(0.0f); accR[i] = (v8f)(0.0f); }

  const float* __restrict__ xrow = x + (size_t)(row0 + m) * IN_FEAT + khalf;
  for (int k0 = 0; k0 < IN_FEAT; k0 += 4) {
    v2f a;
    a.x = xrow[k0];
    a.y = xrow[k0 + 1];
    const int kb = (k0 + khalf) * HC;
#pragma unroll
    for (int tc = 0; tc < 4; ++tc) {
      const int col = tc * 16 + m;
      v2f b, br;
      b.x  = sW [kb + col];
      b.y  = sW [kb + HC + col];
      br.x = sWr[kb + col];
      br.y = sWr[kb + HC + col];
      accW[tc] = __builtin_amdgcn_wmma_f32_16x16x4_f32(false, a, false, b,
                                                       (short)0, accW[tc], false, false);
      accR[tc] = __builtin_amdgcn_wmma_f32_16x16x4_f32(false, a, false, br,
                                                       (short)0, accR[tc], false, false);
    }
  }

  const int rOff = (lane >> 4) << 3;               // 0 or 8
#pragma unroll
  for (int tc = 0; tc < 4; ++tc) {
#pragma unroll
    for (int v = 0; v < 8; ++v) {
      const size_t row = (size_t)(row0 + rOff + v);
      h  [row * HC + tc * 16 + m] = accW[tc][v];
      res[row * HC + tc * 16 + m] = accR[tc][v];
    }
  }
}

// ---------------------------------------------------------------
// Kernel 2: per-node attention logits  al_s[n,h], al_d[n,h]  (b128 loads)
// ---------------------------------------------------------------
__global__ void attn_logits_kernel(const float* __restrict__ h,
                                   const float* __restrict__ a_src,
                                   const float* __restrict__ a_dst,
                                   float* __restrict__ al_s, float* __restrict__ al_d) {
  int n = blockIdx.x * blockDim.x + threadIdx.x;
  if (n >= N_NODES) return;
  const float4* hp = (const float4*)(h + (size_t)n * HC);
  const float4* As = (const float4*)a_src;
  const float4* Ad = (const float4*)a_dst;
#pragma unroll
  for (int hd = 0; hd < HEADS; ++hd) {
    float ss = 0.f, sd = 0.f;
#pragma unroll
    for (int q = 0; q < OUT_F / 4; ++q) {
      float4 hv = hp[hd * 4 + q];
      float4 as = As[hd * 4 + q];
      float4 ad = Ad[hd * 4 + q];
      ss += hv.x * as.x + hv.y * as.y + hv.z * as.z + hv.w * as.w;
      sd += hv.x * ad.x + hv.y * ad.y + hv.z * ad.z + hv.w * ad.w;
    }
    al_s[n * HEADS + hd] = ss;
    al_d[n * HEADS + hd] = sd;
  }
}

__device__ __forceinline__ void edge_ends(const int* __restrict__ ei, int e,
                                          int& src, int& dst) {
  if (e < E_EDGES) { src = ei[e]; dst = ei[E_EDGES + e]; }
  else             { src = dst = e - E_EDGES; }      // self-loop
}

// ---------------------------------------------------------------
// Kernel 3: per-edge leaky_relu logit + per-dst segment max (ordered-uint atomicMax)
// ---------------------------------------------------------------
__global__ void edge_logit_max_kernel(const int* __restrict__ ei,
                                      const float* __restrict__ al_s,
                                      const float* __restrict__ al_d,
                                      float* __restrict__ elogit,
                                      unsigned int* __restrict__ emax) {
  int e = blockIdx.x * blockDim.x + threadIdx.x;
  if (e >= E_TOT) return;
  int src, dst;
  edge_ends(ei, e, src, dst);
#pragma unroll
  for (int hd = 0; hd < HEADS; ++hd) {
    float v = al_s[src * HEADS + hd] + al_d[dst * HEADS + hd];
    v = v > 0.f ? v : NEG_SLOPE * v;                 // leaky_relu
    elogit[(size_t)e * HEADS + hd] = v;
    atomicMax(&emax[dst * HEADS + hd], fenc(v));
  }
}

// ---------------------------------------------------------------
// Kernel 4: segment softmax denominator
// ---------------------------------------------------------------
__global__ void edge_denom_kernel(const int* __restrict__ ei,
                                  const float* __restrict__ elogit,
                                  const unsigned int* __restrict__ emax,
                                  float* __restrict__ denom) {
  int e = blockIdx.x * blockDim.x + threadIdx.x;
  if (e >= E_TOT) return;
  int src, dst;
  edge_ends(ei, e, src, dst);
#pragma unroll
  for (int hd = 0; hd < HEADS; ++hd) {
    float v  = elogit[(size_t)e * HEADS + hd];
    float mx = fdec(emax[dst * HEADS + hd]);
    atomicAdd(&denom[dst * HEADS + hd], __expf(v - mx));
  }
}

// ---------------------------------------------------------------
// Kernel 5: weighted message scatter-add
// one thread per (edge, 4-channel quad): b128 gather of h[src], 4 f32 atomics
// ---------------------------------------------------------------
__global__ void scatter_kernel(const int* __restrict__ ei,
                               const float* __restrict__ elogit,
                               const unsigned int* __restrict__ emax,
                               const float* __restrict__ denom,
                               const float* __restrict__ h,
                               float* __restrict__ sout) {
  long long idx = (long long)blockIdx.x * blockDim.x + threadIdx.x;
  if (idx >= (long long)E_TOT * (HC / 4)) return;
  int e  = (int)(idx >> 4);
  int q  = (int)(idx & 15);          // quad index -> channels q*4 .. q*4+3
  int hd = q >> 2;
  int src, dst;
  edge_ends(ei, e, src, dst);
  float v     = elogit[(size_t)e * HEADS + hd];
  float mx    = fdec(emax[dst * HEADS + hd]);
  float alpha = __expf(v - mx) / denom[dst * HEADS + hd];
  float4 hv   = *(const float4*)(h + (size_t)src * HC + q * 4);
  float* op   = sout + (size_t)dst * HC + q * 4;
  atomicAdd(op + 0, alpha * hv.x);
  atomicAdd(op + 1, alpha * hv.y);
  atomicAdd(op + 2, alpha * hv.z);
  atomicAdd(op + 3, alpha * hv.w);
}

// ---------------------------------------------------------------
// Kernel 6: BatchNorm statistics (sum, sumsq per channel over nodes, bias folded in)
// ---------------------------------------------------------------
__global__ void __launch_bounds__(256)
bn_stats_kernel(const float* __restrict__ sout, const float* __restrict__ bias,
                float* __restrict__ sums /* [0:64]=sum, [64:128]=sumsq */) {
  const int c  = threadIdx.x & 63;
  const int rr = threadIdx.x >> 6;  // 0..3
  const float b = bias[c];
  float s = 0.f, s2 = 0.f;
  for (int n = blockIdx.x * 4 + rr; n < N_NODES; n += gridDim.x * 4) {
    float v = sout[(size_t)n * HC + c] + b;
    s += v; s2 += v * v;
  }
  __shared__ float ls[256], ls2[256];
  ls[threadIdx.x] = s; ls2[threadIdx.x] = s2;
  __syncthreads();
  if (threadIdx.x < 64) {
    s  = ls [c] + ls [64 + c] + ls [128 + c] + ls [192 + c];
    s2 = ls2[c] + ls2[64 + c] + ls2[128 + c] + ls2[192 + c];
    atomicAdd(&sums[c], s);
    atomicAdd(&sums[HC + c], s2);
  }
}

// ---------------------------------------------------------------
// Kernel 7: bias + BN + ELU + residual -> d_out   (float4 in/out)
// ---------------------------------------------------------------
__global__ void final_kernel(const float4* __restrict__ sout4, const float4* __restrict__ bias4,
                             const float* __restrict__ sums, const float4* __restrict__ gamma4,
                             const float4* __restrict__ beta4, const float4* __restrict__ res4,
                             float4* __restrict__ y4) {
  long long idx = (long long)blockIdx.x * blockDim.x + threadIdx.x;
  if (idx >= (long long)N_NODES * (HC / 4)) return;
  const int c4 = (int)(idx & (HC / 4 - 1));
  const float inv_n = 1.0f / (float)N_NODES;
  float4 s  = ((const float4*)sums)[c4];
  float4 s2 = ((const float4*)(sums + HC))[c4];
  float4 b  = bias4[c4];
  float4 g  = gamma4[c4];
  float4 be = beta4[c4];
  float4 v  = sout4[idx];
  float4 r  = res4[idx];
  float4 o;
  {
    float mean = s.x * inv_n, var = s2.x * inv_n - mean * mean;
    float z = (v.x + b.x - mean) * rsqrtf(var + BN_EPS) * g.x + be.x;
    o.x = (z > 0.f ? z : __expf(z) - 1.f) + r.x;
  }
  {
    float mean = s.y * inv_n, var = s2.y * inv_n - mean * mean;
    float z = (v.y + b.y - mean) * rsqrtf(var + BN_EPS) * g.y + be.y;
    o.y = (z > 0.f ? z : __expf(z) - 1.f) + r.y;
  }
  {
    float mean = s.z * inv_n, var = s2.z * inv_n - mean * mean;
    float z = (v.z + b.z - mean) * rsqrtf(var + BN_EPS) * g.z + be.z;
    o.z = (z > 0.f ? z : __expf(z) - 1.f) + r.z;
  }
  {
    float mean = s.w * inv_n, var = s2.w * inv_n - mean * mean;
    float z = (v.w + b.w - mean) * rsqrtf(var + BN_EPS) * g.w + be.w;
    o.w = (z > 0.f ? z : __expf(z) - 1.f) + r.w;
  }
  y4[idx] = o;
}

// ---------------------------------------------------------------
extern "C" void kernel_launch(void* const* d_in, const int* in_sizes, int n_in,
                              void* d_out, int out_size, void* d_ws, size_t ws_size,
                              hipStream_t stream) {
  const float* x      = (const float*)d_in[0];
  const int*   ei     = (const int*)  d_in[1];   // JAX x64 disabled -> int32
  const float* W      = (const float*)d_in[2];
  const float* a_src  = (const float*)d_in[3];
  const float* a_dst  = (const float*)d_in[4];
  const float* bias   = (const float*)d_in[5];
  const float* gamma  = (const float*)d_in[6];
  const float* beta   = (const float*)d_in[7];
  const float* W_res  = (const float*)d_in[8];
  float* y = (float*)d_out;

  // workspace layout (floats)
  float* ws     = (float*)d_ws;
  float* h      = ws;                                   // N*64
  float* res    = h    + (size_t)N_NODES * HC;          // N*64
  float* al_s   = res  + (size_t)N_NODES * HC;          // N*4
  float* al_d   = al_s + (size_t)N_NODES * HEADS;       // N*4
  float* elogit = al_d + (size_t)N_NODES * HEADS;       // E_TOT*4
  unsigned int* emax = (unsigned int*)(elogit + (size_t)E_TOT * HEADS); // N*4
  float* denom  = (float*)(emax + (size_t)N_NODES * HEADS);            // N*4
  float* sout   = denom + (size_t)N_NODES * HEADS;      // N*64
  float* sums   = sout + (size_t)N_NODES * HC;          // 128 (sum | sumsq)

  // 0) init accumulators
  {
    long long tot = (long long)N_NODES * (HC / 4);
    int blocks = (int)((tot + 255) / 256);
    init_kernel<<<blocks, 256, 0, stream>>>((float4*)sout, denom, emax, sums);
  }
  // 1) fused dual GEMM (WMMA f32 16x16x4, async-to-LDS weight staging)
  {
    int rowBlocks = N_NODES / 16;                 // 3125, exact
    int blocks = (rowBlocks + 7) / 8;             // 8 waves per block
    gemm_dual_kernel<<<blocks, 256, 0, stream>>>(x, W, W_res, h, res);
  }
  // 2) attention logits
  attn_logits_kernel<<<(N_NODES + 255) / 256, 256, 0, stream>>>(h, a_src, a_dst, al_s, al_d);
  // 3) edge logits + segment max
  edge_logit_max_kernel<<<(E_TOT + 255) / 256, 256, 0, stream>>>(ei, al_s, al_d, elogit, emax);
  // 4) softmax denominator
  edge_denom_kernel<<<(E_TOT + 255) / 256, 256, 0, stream>>>(ei, elogit, emax, denom);
  // 5) message scatter (edge-quad threads)
  {
    long long tot = (long long)E_TOT * (HC / 4);
    int blocks = (int)((tot + 255) / 256);
    scatter_kernel<<<blocks, 256, 0, stream>>>(ei, elogit, emax, denom, h, sout);
  }
  // 6) BN stats
  bn_stats_kernel<<<512, 256, 0, stream>>>(sout, bias, sums);
  // 7) bias + BN + ELU + residual
  {
    long long tot = (long long)N_NODES * (HC / 4);
    int blocks = (int)((tot + 255) / 256);
    final_kernel<<<blocks, 256, 0, stream>>>((const float4*)sout, (const float4*)bias, sums,
                                             (const float4*)gamma, (const float4*)beta,
                                             (const float4*)res, (float4*)y);
  }
}